// GCNII_13898514169933
// MI455X (gfx1250) — compile-verified
//
#include <hip/hip_runtime.h>
#include <math.h>

typedef __attribute__((ext_vector_type(2))) float v2f;
typedef __attribute__((ext_vector_type(8))) float v8f;

#define N_NODES 50000
#define NEDGES  1600000
#define NFEAT   512
#define NHID    256
#define NCLASS  40
#define NLAYERS 16
#define ALPHA   0.1f
#define LAMDA   0.5

// ---------------------------------------------------------------------------
// Zero a float buffer (float4 granularity)
// ---------------------------------------------------------------------------
__global__ void zero_kernel(float* __restrict__ p, int n4) {
    int i = blockIdx.x * blockDim.x + threadIdx.x;
    if (i < n4) ((float4*)p)[i] = make_float4(0.f, 0.f, 0.f, 0.f);
}

// ---------------------------------------------------------------------------
// support = (1-ALPHA)*hi + ALPHA*h0   (in place on hi)
// ---------------------------------------------------------------------------
__global__ void mix_kernel(float* __restrict__ hi, const float* __restrict__ h0, int n4) {
    int i = blockIdx.x * blockDim.x + threadIdx.x;
    if (i < n4) {
        float4 a = ((float4*)hi)[i];
        float4 b = ((const float4*)h0)[i];
        a.x = (1.0f - ALPHA) * a.x + ALPHA * b.x;
        a.y = (1.0f - ALPHA) * a.y + ALPHA * b.y;
        a.z = (1.0f - ALPHA) * a.z + ALPHA * b.z;
        a.w = (1.0f - ALPHA) * a.w + ALPHA * b.w;
        ((float4*)hi)[i] = a;
    }
}

// ---------------------------------------------------------------------------
// SpMM: out[row] += val * h[col]  — one wave per edge, 8 floats per lane.
// h rows (1KB) and the scatter targets are L2-resident (192MB L2 >> 51MB h).
// ---------------------------------------------------------------------------
__global__ void spmm_kernel(const int* __restrict__ rows, const int* __restrict__ cols,
                            const float* __restrict__ vals, const float* __restrict__ h,
                            float* __restrict__ out, int E) {
    int wid  = threadIdx.x >> 5;
    int lane = threadIdx.x & 31;
    int e = blockIdx.x * (blockDim.x >> 5) + wid;
    if (e >= E) return;
    int   r = rows[e];
    int   c = cols[e];
    float v = vals[e];
    const float4* src = (const float4*)(h + (size_t)c * NHID);
    float* dst = out + (size_t)r * NHID;
#pragma unroll
    for (int half = 0; half < 2; ++half) {
        int idx = lane + half * 32;        // float4 index within the 256-wide row
        float4 x = src[idx];
        atomicAdd(dst + idx * 4 + 0, v * x.x);
        atomicAdd(dst + idx * 4 + 1, v * x.y);
        atomicAdd(dst + idx * 4 + 2, v * x.z);
        atomicAdd(dst + idx * 4 + 3, v * x.w);
    }
}

// ---------------------------------------------------------------------------
// FP32 WMMA GEMM:  C = relu(s_acc*(A@B) + s_res*A[m][n] + bias[n])
//   - wave computes a 32x32 tile via 2x2 V_WMMA_F32_16X16X4_F32 subtiles
//   - s_res path requires K == Nout (layer GEMM reads A as its own residual)
//   - optional dual store (input projection writes h and h0)
// A: [M,K] row-major, B: [K,Nout] row-major, C: [M,Nout]
// ---------------------------------------------------------------------------
__global__ void gemm_wmma_kernel(const float* __restrict__ A, const float* __restrict__ B,
                                 const float* __restrict__ bias,
                                 float* __restrict__ C, float* __restrict__ C2,
                                 int M, int K, int Nout, float s_acc, float s_res) {
    int wid  = threadIdx.x >> 5;
    int lane = threadIdx.x & 31;
    int m0 = blockIdx.x * 32;
    int n0 = (blockIdx.y * (blockDim.x >> 5) + wid) * 32;
    int half = lane >> 4;     // 0: K={0,1} slice, 1: K={2,3} slice (A/B frag layout)
    int lm   = lane & 15;     // row (A) / column (B,D) within the 16-wide tile

    v8f acc00 = {}, acc01 = {}, acc10 = {}, acc11 = {};

    // A row pointers, clamped so the ragged last row-tile stays in bounds
    int ra0 = m0 + lm;        if (ra0 >= M) ra0 = M - 1;
    int ra1 = m0 + 16 + lm;   if (ra1 >= M) ra1 = M - 1;
    const float* Arow0 = A + (size_t)ra0 * K;
    const float* Arow1 = A + (size_t)ra1 * K;

    for (int k = 0; k < K; k += 4) {
        int kk = k + 2 * half;
        // A fragments: contiguous pair {A[m][kk], A[m][kk+1]} -> one b64 load
        v2f a0 = *(const v2f*)(Arow0 + kk);
        v2f a1 = *(const v2f*)(Arow1 + kk);
        // B fragments: B[kk][n], B[kk+1][n]
        const float* Bp = B + (size_t)kk * Nout;
        v2f b0, b1;
        b0.x = Bp[n0 + lm];           b0.y = Bp[Nout + n0 + lm];
        b1.x = Bp[n0 + 16 + lm];      b1.y = Bp[Nout + n0 + 16 + lm];

        acc00 = __builtin_amdgcn_wmma_f32_16x16x4_f32(false, a0, false, b0, (short)0, acc00, false, false);
        acc01 = __builtin_amdgcn_wmma_f32_16x16x4_f32(false, a0, false, b1, (short)0, acc01, false, false);
        acc10 = __builtin_amdgcn_wmma_f32_16x16x4_f32(false, a1, false, b0, (short)0, acc10, false, false);
        acc11 = __builtin_amdgcn_wmma_f32_16x16x4_f32(false, a1, false, b1, (short)0, acc11, false, false);
    }

    // Epilogue. D layout: VGPR r -> (lanes 0-15: M=r, lanes 16-31: M=8+r), N=lane%16.
    v8f accs[2][2] = {{acc00, acc01}, {acc10, acc11}};
#pragma unroll
    for (int i = 0; i < 2; ++i) {
#pragma unroll
        for (int j = 0; j < 2; ++j) {
            int n = n0 + 16 * j + lm;
            float bv = bias ? bias[n] : 0.0f;
#pragma unroll
            for (int r = 0; r < 8; ++r) {
                int m = m0 + 16 * i + 8 * half + r;
                if (m < M) {
                    float res = (s_res != 0.0f) ? A[(size_t)m * K + n] : 0.0f;
                    float val = s_acc * accs[i][j][r] + s_res * res + bv;
                    val = fmaxf(val, 0.0f);
                    C[(size_t)m * Nout + n] = val;
                    if (C2) C2[(size_t)m * Nout + n] = val;
                }
            }
        }
    }
}

// ---------------------------------------------------------------------------
// logits = h @ W_out + b_out ; out = log_softmax(logits, axis=1)
// One 64-thread block per node; h row staged in LDS.
// ---------------------------------------------------------------------------
__global__ void out_softmax_kernel(const float* __restrict__ h, const float* __restrict__ Wout,
                                   const float* __restrict__ bout, float* __restrict__ out) {
    __shared__ float hrow[NHID];
    __shared__ float logit[NCLASS];
    __shared__ float red[2];
    int row = blockIdx.x;
    int t = threadIdx.x;   // 64 threads
    const float* hp = h + (size_t)row * NHID;
    ((float4*)hrow)[t] = ((const float4*)hp)[t];
    __syncthreads();
    if (t < NCLASS) {
        float s = bout[t];
#pragma unroll 4
        for (int k = 0; k < NHID; ++k) s += hrow[k] * Wout[k * NCLASS + t];
        logit[t] = s;
    }
    __syncthreads();
    if (t == 0) {
        float mx = logit[0];
        for (int c = 1; c < NCLASS; ++c) mx = fmaxf(mx, logit[c]);
        float se = 0.0f;
        for (int c = 0; c < NCLASS; ++c) se += expf(logit[c] - mx);
        red[0] = mx;
        red[1] = logf(se);
    }
    __syncthreads();
    if (t < NCLASS) out[(size_t)row * NCLASS + t] = logit[t] - red[0] - red[1];
}

// ---------------------------------------------------------------------------
extern "C" void kernel_launch(void* const* d_in, const int* in_sizes, int n_in,
                              void* d_out, int out_size, void* d_ws, size_t ws_size,
                              hipStream_t stream) {
    const float* x     = (const float*)d_in[0];
    const int*   erow  = (const int*)d_in[1];
    const int*   ecol  = (const int*)d_in[2];
    const float* evalv = (const float*)d_in[3];
    const float* W0    = (const float*)d_in[4];
    const float* b0    = (const float*)d_in[5];
    const float* convW = (const float*)d_in[6];
    const float* Wout  = (const float*)d_in[7];
    const float* bout  = (const float*)d_in[8];
    float* out = (float*)d_out;

    size_t matBytes = (size_t)N_NODES * NHID * sizeof(float);   // 51.2 MB
    float* h  = (float*)d_ws;
    float* h0 = (float*)((char*)d_ws + matBytes);
    float* hi = (float*)((char*)d_ws + 2 * matBytes);

    const int n4 = N_NODES * NHID / 4;
    dim3 gemm_block(128);                       // 4 waves; each owns a 32x32 tile
    dim3 gemm_grid((N_NODES + 31) / 32, NHID / (32 * 4));   // (1563, 2)

    // Input projection: h = h0 = relu(x @ W0 + b0)
    gemm_wmma_kernel<<<gemm_grid, gemm_block, 0, stream>>>(
        x, W0, b0, h, h0, N_NODES, NFEAT, NHID, 1.0f, 0.0f);

    for (int l = 0; l < NLAYERS; ++l) {
        zero_kernel<<<(n4 + 255) / 256, 256, 0, stream>>>(hi, n4);
        spmm_kernel<<<(NEDGES + 7) / 8, 256, 0, stream>>>(erow, ecol, evalv, h, hi, NEDGES);
        mix_kernel<<<(n4 + 255) / 256, 256, 0, stream>>>(hi, h0, n4);
        float theta = (float)log(LAMDA / (double)(l + 1) + 1.0);
        gemm_wmma_kernel<<<gemm_grid, gemm_block, 0, stream>>>(
            hi, convW + (size_t)l * NHID * NHID, nullptr,
            h, nullptr, N_NODES, NHID, NHID, theta, 1.0f - theta);
    }

    out_softmax_kernel<<<N_NODES, 64, 0, stream>>>(h, Wout, bout, out);
}